// RoiAlign_84112639524946
// MI455X (gfx1250) — compile-verified
//
#include <hip/hip_runtime.h>

// RoiAlign (TF crop_and_resize, bilinear, extrapolation_value=0) for MI455X.
//
// Shapes: features [2,64,64,256] f32 (NHWC), rois [2,1000,5] f32,
//         out [2,1000,7,7,256] f32.
//
// Roofline: ~100 MB NT-stored output + ~8 MB L2-resident feature reads
// -> ~4.6 us floor at 23.3 TB/s; compute ~0.2 GFLOP (negligible), so the
// goal is minimal instructions per stored byte. This revision:
//  * weight-product blend (mul + 3 fma per channel) with the 4 bilinear
//    weights computed per-position on the gfx1250 scalar-float SALU
//  * validity mask folded into the weights (no per-lane v_cndmask)
//  * saddr+voffset addressing: per-lane byte offsets hoisted, row bases are
//    scalar adds, store offsets are immediates in the unrolled inner loop
//  * non-temporal b128 stores keep the feature map resident in cache

#define FH 64
#define FW 64
#define FC 256
#define RN 1000
#define NB 2
#define PH 7
#define PW 7

typedef float vf4 __attribute__((ext_vector_type(4)));

__global__ __launch_bounds__(64)
void roialign_cdna5_kernel(const float* __restrict__ feat,
                           const float* __restrict__ rois,
                           float* __restrict__ out) {
    const int r  = blockIdx.x;       // ROI index 0..999
    const int b  = blockIdx.y;       // batch 0..1
    const int c4 = threadIdx.x;      // channel-group 0..63 (4 channels each)

    // Box (y1,x1,y2,x2) = roi[0..3]/1024 — uniform, stays on the SALU.
    const float* __restrict__ roi = rois + ((size_t)b * RN + r) * 5;
    const float inv_dim = 1.0f / 1024.0f;
    const float by1 = roi[0] * inv_dim;
    const float bx1 = roi[1] * inv_dim;
    const float by2 = roi[2] * inv_dim;
    const float bx2 = roi[3] * inv_dim;

    // ---- Hoisted x-direction terms (7 sample columns) -------------------
    // wx0 = (1-lx)*vx, wx1 = lx*vx  (x validity folded in, uniform scalars)
    // off0/off1 = per-lane byte offsets into a feature row (x*1KB + c4*16B)
    float wx0[PW], wx1[PW];
    int   off0[PW], off1[PW];
    const int laneB = c4 * 16;
#pragma unroll
    for (int px = 0; px < PW; ++px) {
        const float tx   = (float)px * (1.0f / (float)(PW - 1));
        const float in_x = (bx1 + (bx2 - bx1) * tx) * (float)(FW - 1);
        const float vx   = ((in_x >= 0.0f) && (in_x <= (float)(FW - 1))) ? 1.0f : 0.0f;
        const float fx   = floorf(in_x);
        const float lx   = in_x - fx;                 // unclipped weight
        wx1[px] = lx * vx;
        wx0[px] = (1.0f - lx) * vx;
        const int x0 = min(max((int)fx, 0), FW - 1);
        const int x1 = min(max((int)ceilf(in_x), 0), FW - 1);
        off0[px] = x0 * (FC * 4) + laneB;
        off1[px] = x1 * (FC * 4) + laneB;
    }

    const char* __restrict__ fb =
        (const char*)feat + (size_t)b * (FH * FW * FC * 4);
    char* __restrict__ ob =
        (char*)out + ((size_t)b * RN + r) * (PH * PW * FC * 4) + laneB;

    // ---- 7x7 sample grid: y terms once per row (scalar) -----------------
    for (int py = 0; py < PH; ++py) {
        const float ty   = (float)py * (1.0f / (float)(PH - 1));
        const float in_y = (by1 + (by2 - by1) * ty) * (float)(FH - 1);
        const float vy   = ((in_y >= 0.0f) && (in_y <= (float)(FH - 1))) ? 1.0f : 0.0f;
        const float fy   = floorf(in_y);
        const float ly   = in_y - fy;
        const float wy1  = ly * vy;
        const float wy0  = (1.0f - ly) * vy;
        const int y0 = min(max((int)fy, 0), FH - 1);
        const int y1 = min(max((int)ceilf(in_y), 0), FH - 1);

        const char* __restrict__ r0 = fb + (size_t)y0 * (FW * FC * 4);
        const char* __restrict__ r1 = fb + (size_t)y1 * (FW * FC * 4);
        char* __restrict__ orow     = ob + (size_t)py * (PW * FC * 4);

#pragma unroll
        for (int px = 0; px < PW; ++px) {
            const vf4 v00 = *(const vf4*)(r0 + off0[px]);   // b128, L2-hot
            const vf4 v01 = *(const vf4*)(r0 + off1[px]);
            const vf4 v10 = *(const vf4*)(r1 + off0[px]);
            const vf4 v11 = *(const vf4*)(r1 + off1[px]);

            // Scalar-unit bilinear weights (mask already folded in; all four
            // are exactly 0 for out-of-box samples -> output is exact 0).
            const float w00 = wy0 * wx0[px];
            const float w01 = wy0 * wx1[px];
            const float w10 = wy1 * wx0[px];
            const float w11 = wy1 * wx1[px];

            // mul + 3 fma per channel
            const vf4 val = v00 * w00 + v01 * w01 + v10 * w10 + v11 * w11;

            // Streaming write-once output: non-temporal b128 store.
            __builtin_nontemporal_store(val, (vf4*)(orow + px * (FC * 4)));
        }
    }
}

extern "C" void kernel_launch(void* const* d_in, const int* in_sizes, int n_in,
                              void* d_out, int out_size, void* d_ws, size_t ws_size,
                              hipStream_t stream) {
    const float* feat = (const float*)d_in[0];   // [2,64,64,256]
    const float* rois = (const float*)d_in[1];   // [2,1000,5]
    float* out        = (float*)d_out;           // [2,1000,7,7,256]

    // One 64-thread block (2 wave32s) per ROI: grid (1000, 2) -> 4000 waves,
    // each issuing 196 independent b128 gathers + 49 NT b128 stores.
    hipLaunchKernelGGL(roialign_cdna5_kernel, dim3(RN, NB), dim3(64), 0, stream,
                       feat, rois, out);
}